// AdaptiveAdjacency_31258771980291
// MI455X (gfx1250) — compile-verified
//
#include <hip/hip_runtime.h>

typedef float v2f __attribute__((ext_vector_type(2)));
typedef float v8f __attribute__((ext_vector_type(8)));

#define B_   32
#define N_   1024
#define FIN_ 64
#define H_   24
#define TILES_M (N_ / 16)          // 64 column tiles
#define ROWS_PER_BLOCK 128         // 8 waves * 16 rows
#define WAVES_PER_BLOCK 8

// ---------------------------------------------------------------------------
// One 16x16 fp32 tile of A = Z Z^T via 6 chained V_WMMA_F32_16X16X4_F32.
// A-operand lane layout (ISA 7.12.2, 32-bit A 16x4): lane l (l16=l&15,h=l>>4):
//   vgpr0 = Z[n0+l16][k0+2h], vgpr1 = Z[n0+l16][k0+2h+1]  -> contiguous float2
// B-operand (4x16, row-striped) mirrors with the column base m0.
// ---------------------------------------------------------------------------
__device__ __forceinline__ v8f tile_mm(const float* __restrict__ Zb,
                                       int n0, int m0, int l16, int half) {
  v8f c = {0.f, 0.f, 0.f, 0.f, 0.f, 0.f, 0.f, 0.f};
  const float* arow = Zb + (n0 + l16) * H_ + 2 * half;
  const float* brow = Zb + (m0 + l16) * H_ + 2 * half;
#pragma unroll
  for (int s = 0; s < 6; ++s) {
    v2f av = *(const v2f*)(arow + 4 * s);
    v2f bv = *(const v2f*)(brow + 4 * s);
    c = __builtin_amdgcn_wmma_f32_16x16x4_f32(
        /*neg_a=*/false, av, /*neg_b=*/false, bv,
        /*c_mod=*/(short)0, c, /*reuse_a=*/false, /*reuse_b=*/false);
  }
  return c;
}

// ---------------------------------------------------------------------------
// Kernel 1: Z = relu(xt @ W + b)   (B*N x H)
// ---------------------------------------------------------------------------
__global__ void zrelu_k(const float* __restrict__ xt, const float* __restrict__ W,
                        const float* __restrict__ bias, float* __restrict__ Z) {
  __shared__ float Ws[FIN_ * H_];
  for (int i = threadIdx.x; i < FIN_ * H_; i += blockDim.x) Ws[i] = W[i];
  __syncthreads();
  int gid = blockIdx.x * blockDim.x + threadIdx.x;
  int h = gid & 31;
  int row = gid >> 5;
  if (row >= B_ * N_ || h >= H_) return;
  const float* xr = xt + (size_t)row * FIN_;
  float acc = bias[h];
#pragma unroll
  for (int f = 0; f < FIN_; ++f) acc = fmaf(xr[f], Ws[f * H_ + h], acc);
  Z[(size_t)row * H_ + h] = fmaxf(acc, 0.0f);
}

// ---------------------------------------------------------------------------
// Kernel 2: online row softmax stats. A symmetric => col stats == row stats.
// Each wave owns 16 rows, sweeps all 64 column tiles.
// ---------------------------------------------------------------------------
__global__ void rowstats_k(const float* __restrict__ Z,
                           float* __restrict__ rmax, float* __restrict__ rrsum) {
  int wave = threadIdx.x >> 5, lane = threadIdx.x & 31;
  int l16 = lane & 15, half = lane >> 4;
  int b = blockIdx.x;
  int n0 = blockIdx.y * ROWS_PER_BLOCK + wave * 16;
  const float* Zb = Z + (size_t)b * N_ * H_;

  float rm[8], rs[8];
#pragma unroll
  for (int r = 0; r < 8; ++r) { rm[r] = -3.402823466e38f; rs[r] = 0.f; }

  for (int mt = 0; mt < TILES_M; ++mt) {
    v8f c = tile_mm(Zb, n0, mt * 16, l16, half);
#pragma unroll
    for (int r = 0; r < 8; ++r) {
      float v = c[r];
      float mn = fmaxf(rm[r], v);
      rs[r] = rs[r] * __expf(rm[r] - mn) + __expf(v - mn);
      rm[r] = mn;
    }
  }
  // merge across the 16-lane column group (stays inside each wave half)
#pragma unroll
  for (int mask = 1; mask < 16; mask <<= 1) {
#pragma unroll
    for (int r = 0; r < 8; ++r) {
      float om = __shfl_xor(rm[r], mask, 32);
      float os = __shfl_xor(rs[r], mask, 32);
      float mn = fmaxf(rm[r], om);
      rs[r] = rs[r] * __expf(rm[r] - mn) + os * __expf(om - mn);
      rm[r] = mn;
    }
  }
  if (l16 == 0) {
#pragma unroll
    for (int r = 0; r < 8; ++r) {
      int n = n0 + r + 8 * half;
      rmax[b * N_ + n] = rm[r];
      rrsum[b * N_ + n] = 1.0f / rs[r];
    }
  }
}

// ---------------------------------------------------------------------------
// Kernel 3: t[n] = sum_m exp(A[n,m]-rmax[m]) * rrsum[m]  (col-softmax row sum)
// ---------------------------------------------------------------------------
__global__ void tsum_k(const float* __restrict__ Z, const float* __restrict__ rmax,
                       const float* __restrict__ rrsum, float* __restrict__ t) {
  __shared__ float smax[N_];
  __shared__ float srr[N_];
  int b = blockIdx.x;
  for (int i = threadIdx.x; i < N_; i += blockDim.x) {
    smax[i] = rmax[b * N_ + i];
    srr[i]  = rrsum[b * N_ + i];
  }
  __syncthreads();

  int wave = threadIdx.x >> 5, lane = threadIdx.x & 31;
  int l16 = lane & 15, half = lane >> 4;
  int n0 = blockIdx.y * ROWS_PER_BLOCK + wave * 16;
  const float* Zb = Z + (size_t)b * N_ * H_;

  float acc[8];
#pragma unroll
  for (int r = 0; r < 8; ++r) acc[r] = 0.f;

  for (int mt = 0; mt < TILES_M; ++mt) {
    int m0 = mt * 16;
    float cm = smax[m0 + l16];
    float cr = srr[m0 + l16];
    v8f c = tile_mm(Zb, n0, m0, l16, half);
#pragma unroll
    for (int r = 0; r < 8; ++r) acc[r] += __expf(c[r] - cm) * cr;
  }
#pragma unroll
  for (int mask = 1; mask < 16; mask <<= 1) {
#pragma unroll
    for (int r = 0; r < 8; ++r) acc[r] += __shfl_xor(acc[r], mask, 32);
  }
  if (l16 == 0) {
#pragma unroll
    for (int r = 0; r < 8; ++r) t[b * N_ + n0 + r + 8 * half] = acc[r];
  }
}

// ---------------------------------------------------------------------------
// Kernel 4: dinv = (clip(1.5 + 0.5*t, 1e-6))^-1/2
// ---------------------------------------------------------------------------
__global__ void dinv_k(const float* __restrict__ t, float* __restrict__ dinv) {
  int i = blockIdx.x * blockDim.x + threadIdx.x;
  if (i < B_ * N_) {
    float deg = fmaxf(1.5f + 0.5f * t[i], 1e-6f);
    dinv[i] = rsqrtf(deg);
  }
}

// ---------------------------------------------------------------------------
// Kernel 5: recompute tiles, apply symmetric softmax + self-loop + Kipf norm,
// stream 128 MB out. This pass is the HBM-write roofline floor (~5.5 us).
// ---------------------------------------------------------------------------
__global__ void out_k(const float* __restrict__ Z, const float* __restrict__ rmax,
                      const float* __restrict__ rrsum, const float* __restrict__ dinv,
                      float* __restrict__ out) {
  __shared__ float smax[N_];
  __shared__ float srr[N_];
  __shared__ float sdi[N_];
  int b = blockIdx.x;
  for (int i = threadIdx.x; i < N_; i += blockDim.x) {
    smax[i] = rmax[b * N_ + i];
    srr[i]  = rrsum[b * N_ + i];
    sdi[i]  = dinv[b * N_ + i];
  }
  __syncthreads();

  int wave = threadIdx.x >> 5, lane = threadIdx.x & 31;
  int l16 = lane & 15, half = lane >> 4;
  int n0 = blockIdx.y * ROWS_PER_BLOCK + wave * 16;
  const float* Zb = Z + (size_t)b * N_ * H_;
  float* outb = out + (size_t)b * N_ * N_;

  float nm[8], nr[8], nd[8];
#pragma unroll
  for (int r = 0; r < 8; ++r) {
    int n = n0 + r + 8 * half;
    nm[r] = smax[n]; nr[r] = srr[n]; nd[r] = sdi[n];
  }

  for (int mt = 0; mt < TILES_M; ++mt) {
    int m0 = mt * 16;
    int m = m0 + l16;
    float cm = smax[m];
    float cr = srr[m];
    float cd = sdi[m];
    v8f c = tile_mm(Zb, n0, m0, l16, half);
#pragma unroll
    for (int r = 0; r < 8; ++r) {
      int n = n0 + r + 8 * half;
      float v = c[r];
      float e1 = __expf(v - nm[r]) * nr[r];
      float e2 = __expf(v - cm) * cr;
      float o = 0.5f * (e1 + e2) + ((n == m) ? 1.0f : 0.0f);
      o *= nd[r] * cd;
      outb[(size_t)n * N_ + m] = o;
    }
  }
}

// ---------------------------------------------------------------------------
extern "C" void kernel_launch(void* const* d_in, const int* in_sizes, int n_in,
                              void* d_out, int out_size, void* d_ws, size_t ws_size,
                              hipStream_t stream) {
  const float* xt   = (const float*)d_in[0];
  const float* W    = (const float*)d_in[1];
  const float* bias = (const float*)d_in[2];
  float* out = (float*)d_out;

  float* ws    = (float*)d_ws;
  float* Z     = ws;                      // B*N*H      = 786432 f32
  float* rmax  = Z + (size_t)B_ * N_ * H_;// B*N        =  32768 f32
  float* rrsum = rmax + B_ * N_;
  float* t     = rrsum + B_ * N_;
  float* dinv  = t + B_ * N_;             // total ~3.5 MB of workspace

  // 1) Z = relu(xt @ W + b)
  zrelu_k<<<(B_ * N_ * 32) / 256, 256, 0, stream>>>(xt, W, bias, Z);

  dim3 grid(B_, N_ / ROWS_PER_BLOCK);   // (32, 8) blocks of 8 waves

  // 2) row softmax stats (== col stats by symmetry)
  rowstats_k<<<grid, 256, 0, stream>>>(Z, rmax, rrsum);

  // 3) t = row-sums of the column softmax
  tsum_k<<<grid, 256, 0, stream>>>(Z, rmax, rrsum, t);

  // 4) Kipf degree^-1/2
  dinv_k<<<(B_ * N_ + 255) / 256, 256, 0, stream>>>(t, dinv);

  // 5) fused final pass -> 128 MB write
  out_k<<<grid, 256, 0, stream>>>(Z, rmax, rrsum, dinv, out);
}